// Queue_59081570124277
// MI455X (gfx1250) — compile-verified
//
#include <hip/hip_runtime.h>
#include <hip/hip_bf16.h>

// Problem constants (reference: T=512 steps/slots, B=64 batch, E=512 embed)
#define TT 512
#define BB 64
#define EE 512

typedef __attribute__((ext_vector_type(2))) float v2f;
typedef __attribute__((ext_vector_type(8))) float v8f;
typedef int v4i_ls __attribute__((vector_size(16)));  // builtin's pointee type

// ---------------------------------------------------------------------------
// CDNA5 async global->LDS helpers (ASYNCcnt-tracked DMA; guarded fallback)
// ---------------------------------------------------------------------------
#if __has_builtin(__builtin_amdgcn_global_load_async_to_lds_b128)
#define HAS_ASYNC_LDS 1
#else
#define HAS_ASYNC_LDS 0
#endif

__device__ __forceinline__ void copy_g2l_b128(const float* __restrict__ g,
                                              float* l) {
#if HAS_ASYNC_LDS
  __builtin_amdgcn_global_load_async_to_lds_b128(
      (v4i_ls*)g, (v4i_ls*)l, /*offset=*/0, /*cpol=*/0);
#else
  *reinterpret_cast<float4*>(l) = *reinterpret_cast<const float4*>(g);
#endif
}

template <int N>
__device__ __forceinline__ void wait_asynccnt() {
#if __has_builtin(__builtin_amdgcn_s_wait_asynccnt)
  __builtin_amdgcn_s_wait_asynccnt(N);
#else
  asm volatile("s_wait_asynccnt %0" ::"n"(N));
#endif
}

// ---------------------------------------------------------------------------
// Kernel 1: queue-strength recurrence -> coefficients C[b][t][i] (d_ws)
// One block per batch, thread i == slot i. One exclusive block scan per step
// (the post-push cumsum of step t is the pre-pop cumsum of step t+1).
// ---------------------------------------------------------------------------
__global__ __launch_bounds__(TT) void queue_coeffs_kernel(
    const float* __restrict__ U, const float* __restrict__ D,
    float* __restrict__ C) {
  const int b    = blockIdx.x;
  const int i    = threadIdx.x;       // slot
  const int lane = i & 31;
  const int wid  = i >> 5;            // 16 waves of 32

  __shared__ float sU[TT];
  __shared__ float sD[TT];
  __shared__ float waveSum[16];
  __shared__ float waveOff[16];

  sU[i] = U[(size_t)i * BB + b];
  sD[i] = D[(size_t)i * BB + b];
  __syncthreads();

  float s = 0.0f;      // strength of slot i
  float excl = 0.0f;   // sum of strengths in slots < i (current state)
  float* crow = C + (size_t)b * TT * TT + i;

  for (int t = 0; t < TT; ++t) {
    const float u = sU[t];
    const float d = sD[t];
    // pop
    float w = fmaxf(u - excl, 0.0f);
    s = fmaxf(s - w, 0.0f);
    // push d into slot t
    if (i == t) s += d;
    // exclusive block scan of s across 512 slots
    float v = s;
#pragma unroll
    for (int off = 1; off < 32; off <<= 1) {
      float n = __shfl_up(v, off, 32);
      if (lane >= off) v += n;
    }
    if (lane == 31) waveSum[wid] = v;
    __syncthreads();
    if (wid == 0) {
      float ws = (lane < 16) ? waveSum[lane] : 0.0f;
#pragma unroll
      for (int off = 1; off < 16; off <<= 1) {
        float n = __shfl_up(ws, off, 32);
        if (lane >= off) ws += n;
      }
      if (lane < 16) waveOff[lane] = ws - waveSum[lane];  // exclusive offsets
    }
    __syncthreads();
    const float incl = v + waveOff[wid];
    excl = incl - s;  // exclusive cumsum of post-push state (reused next step)
    // read coefficients
    crow[(size_t)t * TT] = fminf(s, fmaxf(1.0f - excl, 0.0f));
  }
}

// ---------------------------------------------------------------------------
// Kernel 2: batched GEMM  out[t,b,e] = sum_i C[b][t][i] * V[i,b,e]
// 128x128 output tile / 256-thread block; 8 waves in 2x4 grid, each wave a
// 64x32 region = 4x2 grid of 16x16 f32 accumulators via V_WMMA_F32_16X16X4_F32.
// K staged through double-buffered LDS with GLOBAL_LOAD_ASYNC_TO_LDS_B128:
// stage i+1's DMA overlaps stage i's 32 WMMAs; s_wait_asynccnt(4) + barrier
// (4 async ops per stage per wave, in-order completion) gates consumption.
// ---------------------------------------------------------------------------
#define KC 16
#define ASTRIDE (KC + 4)     // 20 dwords: f32x2-aligned, conflict-free frag reads
#define BSTRIDE (128 + 8)    // 136 dwords: 2*136 % 64 == 16 -> half-waves split banks

__global__ __launch_bounds__(256) void queue_gemm_kernel(
    const float* __restrict__ Cws, const float* __restrict__ V,
    float* __restrict__ out) {
  __shared__ __align__(16) float As[2][128 * ASTRIDE];  // [buf][row][k]
  __shared__ __align__(16) float Bs[2][KC * BSTRIDE];   // [buf][k][col]

  const int b   = blockIdx.z;
  const int bm  = blockIdx.y * 128;
  const int bn  = blockIdx.x * 128;
  const int tid = threadIdx.x;
  const int lane  = tid & 31;
  const int wid   = tid >> 5;          // 0..7
  const int wm    = (wid >> 2) * 64;   // wave M offset: 0 / 64
  const int wn    = (wid & 3) * 32;    // wave N offset: 0/32/64/96
  const int lhalf = lane >> 4;         // 0: K=k,k+1   1: K=k+2,k+3
  const int l16   = lane & 15;

  v8f acc[4][2];
#pragma unroll
  for (int ms = 0; ms < 4; ++ms)
#pragma unroll
    for (int ns = 0; ns < 2; ++ns) acc[ms][ns] = (v8f)0.0f;

  const float* Abase = Cws + ((size_t)b * TT + bm) * TT;    // C[b][bm+..][..]
  const float* Vbase = V + (size_t)b * EE + bn;             // V[i][b][bn+..]

  // Per-thread staging coordinates (2 float4 of A + 2 float4 of B each)
  const int ar0 = tid >> 2, ac0 = (tid & 3) << 2;            // A chunk 0
  const int ar1 = (tid + 256) >> 2, ac1 = ac0;               // A chunk 1
  const int bk0 = tid >> 5, bc0 = (tid & 31) << 2;           // B chunk 0
  const int bk1 = (tid + 256) >> 5, bc1 = bc0;               // B chunk 1

  auto stage = [&](int buf, int k0) {
    copy_g2l_b128(Abase + (size_t)ar0 * TT + k0 + ac0,
                  &As[buf][ar0 * ASTRIDE + ac0]);
    copy_g2l_b128(Abase + (size_t)ar1 * TT + k0 + ac1,
                  &As[buf][ar1 * ASTRIDE + ac1]);
    copy_g2l_b128(Vbase + (size_t)(k0 + bk0) * (BB * EE) + bc0,
                  &Bs[buf][bk0 * BSTRIDE + bc0]);
    copy_g2l_b128(Vbase + (size_t)(k0 + bk1) * (BB * EE) + bc1,
                  &Bs[buf][bk1 * BSTRIDE + bc1]);
  };

  // Prologue: prime both buffers
  stage(0, 0);
  stage(1, KC);

  int buf = 0;
  for (int k0 = 0; k0 < TT; k0 += KC) {
    // Ensure this stage's 4 async ops landed (in-order per wave): if another
    // stage is still in flight allow its 4 to remain outstanding.
    if (k0 + KC < TT) wait_asynccnt<4>();
    else              wait_asynccnt<0>();
    __syncthreads();

    const float* Al = As[buf];
    const float* Bl = Bs[buf];
#pragma unroll
    for (int kk = 0; kk < KC; kk += 4) {
      const int ksel = kk + lhalf * 2;
      // A fragments: 16x4 f32; lane row = l16, two K values per lane
      v2f afrag[4];
#pragma unroll
      for (int ms = 0; ms < 4; ++ms) {
        const int row = wm + ms * 16 + l16;
        afrag[ms] = *reinterpret_cast<const v2f*>(&Al[row * ASTRIDE + ksel]);
      }
      // B fragments: 4x16 f32; VGPR0 = rows K=ksel/ksel+2, VGPR1 = +1
      v2f bfrag[2];
#pragma unroll
      for (int ns = 0; ns < 2; ++ns) {
        const int col = wn + ns * 16 + l16;
        bfrag[ns].x = Bl[ksel * BSTRIDE + col];
        bfrag[ns].y = Bl[(ksel + 1) * BSTRIDE + col];
      }
#pragma unroll
      for (int ms = 0; ms < 4; ++ms)
#pragma unroll
        for (int ns = 0; ns < 2; ++ns)
          acc[ms][ns] = __builtin_amdgcn_wmma_f32_16x16x4_f32(
              /*neg_a=*/false, afrag[ms], /*neg_b=*/false, bfrag[ns],
              /*c_mod=*/(short)0, acc[ms][ns],
              /*reuse_a=*/false, /*reuse_b=*/false);
    }
    __syncthreads();  // all waves done reading buf before DMA overwrites it

    if (k0 + 2 * KC < TT) stage(buf, k0 + 2 * KC);
    buf ^= 1;
  }

  // Store D tiles: VGPR r holds (M = 8*lhalf + r, N = l16) of each 16x16 tile
#pragma unroll
  for (int ms = 0; ms < 4; ++ms) {
#pragma unroll
    for (int ns = 0; ns < 2; ++ns) {
      const int col = bn + wn + ns * 16 + l16;
#pragma unroll
      for (int r = 0; r < 8; ++r) {
        const int row = bm + wm + ms * 16 + lhalf * 8 + r;
        out[((size_t)row * BB + b) * EE + col] = acc[ms][ns][r];
      }
    }
  }
}

// ---------------------------------------------------------------------------
extern "C" void kernel_launch(void* const* d_in, const int* in_sizes, int n_in,
                              void* d_out, int out_size, void* d_ws,
                              size_t ws_size, hipStream_t stream) {
  const float* V = (const float*)d_in[0];  // [T, B, E]
  const float* U = (const float*)d_in[1];  // [T, B]
  const float* D = (const float*)d_in[2];  // [T, B]
  float* out = (float*)d_out;              // [T, B, E]
  float* Cws = (float*)d_ws;               // needs T*T*B*4 = 64 MiB scratch

  (void)in_sizes; (void)n_in; (void)out_size; (void)ws_size;

  queue_coeffs_kernel<<<dim3(BB), dim3(TT), 0, stream>>>(U, D, Cws);
  queue_gemm_kernel<<<dim3(EE / 128, TT / 128, BB), dim3(256), 0, stream>>>(
      Cws, V, out);
}